// GLM_64355789963658
// MI455X (gfx1250) — compile-verified
//
#include <hip/hip_runtime.h>

#define DINL __device__ __forceinline__

typedef __attribute__((ext_vector_type(2))) float v2f;
typedef __attribute__((ext_vector_type(8))) float v8f;

constexpr int BATCH  = 16;
constexpr int NBLK   = 4096;
constexpr int HDIM   = 128;
constexpr int NEDGE  = 65536;
constexpr int NLAYER = 2;
constexpr int NROWS  = BATCH * NBLK;   // 65536 GEMM rows
constexpr int KB     = 64;             // K panel staged in LDS
constexpr float LN_EPS = 1e-5f;

// ---------------------------------------------------------------- A loaders
struct ALoadPlain {
  const float* A; int ld;
  DINL v2f load2(int r, int k) const {
    float2 t = *(const float2*)&A[(size_t)r * ld + k];
    v2f o; o.x = t.x; o.y = t.y; return o;
  }
};

struct ALoadConcat {           // k < H -> A0, else A1 (gate input concat)
  const float* A0; const float* A1;
  DINL v2f load2(int r, int k) const {
    const float* p = (k < HDIM) ? &A0[(size_t)r * HDIM + k]
                                : &A1[(size_t)r * HDIM + (k - HDIM)];
    float2 t = *(const float2*)p;
    v2f o; o.x = t.x; o.y = t.y; return o;
  }
};

struct ALoadLN {               // fused layernorm on the fly
  const float* X; const float* mu; const float* rs;
  const float* sc; const float* bi;
  DINL v2f load2(int r, int k) const {
    float2 x = *(const float2*)&X[(size_t)r * HDIM + k];
    float2 s = *(const float2*)&sc[k];
    float2 b = *(const float2*)&bi[k];
    float m = mu[r], ir = rs[r];
    v2f o;
    o.x = (x.x - m) * ir * s.x + b.x;
    o.y = (x.y - m) * ir * s.y + b.y;
    return o;
  }
};

// ---------------------------------------------------------------- epilogues
struct EpiElu {                // out = elu(acc + bias)
  const float* bias; float* out; int ldo;
  DINL void operator()(int r, int c, float v) const {
    v += bias[c];
    out[(size_t)r * ldo + c] = v > 0.f ? v : expm1f(v);
  }
};

struct EpiGate {               // x = bt + sigmoid(acc + bias) * msg
  const float* bias; const float* bt; const float* msg; float* out;
  DINL void operator()(int r, int c, float v) const {
    const size_t i = (size_t)r * HDIM + c;
    const float g = 1.f / (1.f + __expf(-(v + bias[c])));
    out[i] = bt[i] + g * msg[i];
  }
};

struct EpiRes {                // out = xin + acc + bias
  const float* bias; const float* xin; float* out;
  DINL void operator()(int r, int c, float v) const {
    const size_t i = (size_t)r * HDIM + c;
    out[i] = xin[i] + v + bias[c];
  }
};

// ------------------------------------------------- fp32 WMMA GEMM (fused)
// 256 threads = 8 waves; each wave owns a 16-row x 128-col slab.
// B staged in LDS in WMMA-fragment order (32-lane x 2-float blocks per
// (ks, ntile)), so each B operand pair is one aligned ds_load_b64 and the
// compiler fuses two fragments per ds_load_2addr_b64. Staging uses the
// CDNA5 async DMA path: global_load_async_to_lds_b32 writes the swizzled
// fragment slots directly (per-lane LDS dest), tracked by ASYNCcnt.
template <int K, int N, class AL, class EP>
__global__ __launch_bounds__(256) void gemm_kernel(const AL al,
                                                   const float* __restrict__ Bmat,
                                                   const EP ep) {
  __shared__ float sB[KB * 128];       // 32 KB, fragment-major
  const int tid  = threadIdx.x;
  const int wv   = tid >> 5;
  const int lane = tid & 31;
  const int hf   = lane >> 4;          // half-wave select (K split)
  const int l16  = lane & 15;
  const int row0 = blockIdx.x * 128 + wv * 16;
  const float* sfrag = sB + lane * 2;  // per-lane fragment base

  // staging decomposition: column within tile, K-pair half, N tile
  const int st_n15 = tid & 15;
  const int st_hfk = (tid >> 4) & 1;
  const int st_nt  = tid >> 5;
  // LDS byte offset of this thread's fragment slot family (k0 = 0, ks = 0)
  const unsigned ldsB =
      (unsigned)(uintptr_t)&sB[(st_nt * 32 + st_hfk * 16 + st_n15) * 2];

  for (int nb = 0; nb < N; nb += 128) {
    v8f acc[8];
#pragma unroll
    for (int i = 0; i < 8; ++i)
#pragma unroll
      for (int j = 0; j < 8; ++j) acc[i][j] = 0.f;

#pragma unroll
    for (int kp = 0; kp < K; kp += KB) {
      __syncthreads();   // all waves done reading the previous panel
      // async DMA: element (kp+4ks+2hfk+k0, nb+nt*16+n15) -> fragment slot
      {
        const float* gB = Bmat + (size_t)(kp + st_hfk * 2) * N + nb +
                          st_nt * 16 + st_n15;
#pragma unroll
        for (int ks = 0; ks < KB / 4; ++ks) {
#pragma unroll
          for (int k0 = 0; k0 < 2; ++k0) {
            const float* g = gB + (size_t)(ks * 4 + k0) * N;
            const unsigned l = ldsB + (unsigned)(ks * 2048 + k0 * 4);
            asm volatile("global_load_async_to_lds_b32 %0, %1, off"
                         :: "v"(l), "v"(g) : "memory");
          }
        }
        asm volatile("s_wait_asynccnt 0" ::: "memory");
      }
      __syncthreads();   // everyone's async writes now visible in LDS

      // A fragments for this K panel (reused across all 8 N tiles)
      v2f afr[KB / 4];
#pragma unroll
      for (int ks = 0; ks < KB / 4; ++ks)
        afr[ks] = al.load2(row0 + l16, kp + ks * 4 + hf * 2);

#pragma unroll
      for (int ks = 0; ks < KB / 4; ++ks) {
#pragma unroll
        for (int n = 0; n < 8; ++n) {
          const v2f bfr = *(const v2f*)(sfrag + (ks * 8 + n) * 64);
          acc[n] = __builtin_amdgcn_wmma_f32_16x16x4_f32(
              false, afr[ks], false, bfr, (short)0, acc[n], false, false);
        }
      }
    }

    // epilogue: C/D layout -> (row0 + r + hf*8, nb + n*16 + l16)
#pragma unroll
    for (int n = 0; n < 8; ++n)
#pragma unroll
      for (int r = 0; r < 8; ++r)
        ep(row0 + r + hf * 8, nb + n * 16 + l16, acc[n][r]);
  }
}

// ---------------------------------------------------------- edge weights
__global__ __launch_bounds__(256) void k_edge(const float* __restrict__ estr,
                                              const float* __restrict__ ctemb,
                                              const int* __restrict__ ctype,
                                              const int* __restrict__ esrc,
                                              const int* __restrict__ edst,
                                              const unsigned char* __restrict__ act,
                                              float* __restrict__ ew) {
  const int e = blockIdx.x * 256 + threadIdx.x;
  const int b = blockIdx.y;
  const float s = 1.f / (1.f + __expf(-estr[e]));
  const float w = s * ctemb[(size_t)ctype[b] * NEDGE + e];
  const bool on = act[(size_t)b * NBLK + esrc[e]] && act[(size_t)b * NBLK + edst[e]];
  ew[(size_t)b * NEDGE + e] = on ? w : 0.f;
}

// --------------------------------------------- CSR-by-destination build
// (edge_src/edge_dst are batch-independent -> built once, reused by all B)
__global__ __launch_bounds__(256) void k_hist(const int* __restrict__ edst,
                                              int* __restrict__ counts) {
  const int e = blockIdx.x * 256 + threadIdx.x;
  atomicAdd(&counts[edst[e]], 1);
}

__global__ __launch_bounds__(256) void k_scan(const int* __restrict__ counts,
                                              int* __restrict__ offs) {
  __shared__ int sh[256];
  __shared__ int carrySh;
  const int tid = threadIdx.x;
  if (tid == 0) carrySh = 0;
  __syncthreads();
  for (int base = 0; base < NBLK; base += 256) {
    int v = counts[base + tid];
    const int orig = v;
    for (int o = 1; o < 256; o <<= 1) {
      sh[tid] = v;
      __syncthreads();
      const int add = (tid >= o) ? sh[tid - o] : 0;
      __syncthreads();
      v += add;
    }
    const int carry = carrySh;
    offs[base + tid] = carry + v - orig;     // exclusive prefix
    __syncthreads();
    if (tid == 255) carrySh = carry + v;
    __syncthreads();
  }
  if (tid == 0) offs[NBLK] = NEDGE;
}

__global__ __launch_bounds__(256) void k_fill(const int* __restrict__ edst,
                                              const int* __restrict__ offs,
                                              int* __restrict__ cursor,
                                              int* __restrict__ list) {
  const int e = blockIdx.x * 256 + threadIdx.x;
  const int d = edst[e];
  const int p = atomicAdd(&cursor[d], 1);
  list[offs[d] + p] = e;
}

// canonical (ascending edge id) segment order -> deterministic FP sums
__global__ __launch_bounds__(256) void k_sortseg(const int* __restrict__ offs,
                                                 int* __restrict__ list) {
  const int d = blockIdx.x * 256 + threadIdx.x;
  const int s = offs[d], t = offs[d + 1];
  for (int i = s + 1; i < t; ++i) {
    const int v = list[i];
    int j = i - 1;
    while (j >= s && list[j] > v) { list[j + 1] = list[j]; --j; }
    list[j + 1] = v;
  }
}

// ----------------------- atomic-free message pass: gather + reg accumulate
// one wave per (batch, dst); float4 per lane; single plain store
__global__ __launch_bounds__(256) void k_msg(const float* __restrict__ ew,
                                             const float* __restrict__ hroute,
                                             const int* __restrict__ esrc,
                                             const int* __restrict__ offs,
                                             const int* __restrict__ list,
                                             float* __restrict__ msgs) {
  const int lane = threadIdx.x & 31;
  const int d = blockIdx.x * 8 + (threadIdx.x >> 5);
  const int b = blockIdx.y;
  const int s0 = offs[d], s1 = offs[d + 1];
  float4 acc = {0.f, 0.f, 0.f, 0.f};
  for (int i = s0; i < s1; ++i) {
    const int e = list[i];
    const float w = ew[(size_t)b * NEDGE + e];
    if (w != 0.f) {
      const float4 h =
          *(const float4*)&hroute[((size_t)b * NBLK + esrc[e]) * HDIM + lane * 4];
      acc.x += w * h.x;
      acc.y += w * h.y;
      acc.z += w * h.z;
      acc.w += w * h.w;
    }
  }
  *(float4*)&msgs[((size_t)b * NBLK + d) * HDIM + lane * 4] = acc;
}

// ------------------------------------------------------- LN row statistics
__global__ __launch_bounds__(256) void k_lnstats(const float* __restrict__ x,
                                                 float* __restrict__ stats) {
  const int lane = threadIdx.x & 31;
  const int row  = blockIdx.x * 8 + (threadIdx.x >> 5);
  const float4 v = *(const float4*)&x[(size_t)row * HDIM + lane * 4];
  float s  = v.x + v.y + v.z + v.w;
  float ss = v.x * v.x + v.y * v.y + v.z * v.z + v.w * v.w;
#pragma unroll
  for (int o = 16; o > 0; o >>= 1) {
    s  += __shfl_down(s, o, 32);
    ss += __shfl_down(ss, o, 32);
  }
  if (lane == 0) {
    const float mu  = s * (1.f / HDIM);
    const float var = ss * (1.f / HDIM) - mu * mu;
    stats[row] = mu;
    stats[NROWS + row] = rsqrtf(var + LN_EPS);
  }
}

// ------------------------------------------------------------------ driver
extern "C" void kernel_launch(void* const* d_in, const int* in_sizes, int n_in,
                              void* d_out, int out_size, void* d_ws, size_t ws_size,
                              hipStream_t stream) {
  const float* bt    = (const float*)d_in[0];
  const float* estr  = (const float*)d_in[1];
  const float* ctemb = (const float*)d_in[2];
  const float* Wproj = (const float*)d_in[3];
  const float* bproj = (const float*)d_in[4];
  const float* Wgate = (const float*)d_in[5];
  const float* bgate = (const float*)d_in[6];
  const float* lnsc  = (const float*)d_in[7];
  const float* lnbi  = (const float*)d_in[8];
  const float* W1    = (const float*)d_in[9];
  const float* b1    = (const float*)d_in[10];
  const float* W2    = (const float*)d_in[11];
  const float* b2    = (const float*)d_in[12];
  const int*   ctype = (const int*)d_in[13];
  const int*   esrc  = (const int*)d_in[14];
  const int*   edst  = (const int*)d_in[15];
  const unsigned char* act = (const unsigned char*)d_in[16];

  float* out_x  = (float*)d_out;                       // (B, NB, H)
  float* out_ew = out_x + (size_t)NROWS * HDIM;        // (B, E)

  float* x     = (float*)d_ws;                         // R*H
  float* msgs  = x + (size_t)NROWS * HDIM;             // R*H
  float* stats = msgs + (size_t)NROWS * HDIM;          // 2*R (mu | rsigma)
  float* big   = stats + 2 * (size_t)NROWS;            // R*2H (h_route / hidden)
  int*   offs  = (int*)(big + (size_t)NROWS * 2 * HDIM);  // NBLK+1
  int*   tmp   = offs + (NBLK + 1);                    // counts, then cursor
  int*   list  = tmp + NBLK;                           // NEDGE edge ids

  // 1. edge weights straight into output slice (also feeds message pass)
  k_edge<<<dim3(NEDGE / 256, BATCH), 256, 0, stream>>>(estr, ctemb, ctype, esrc,
                                                       edst, act, out_ew);

  // 2. h_route = elu(bt @ W_proj + b_proj)
  gemm_kernel<HDIM, HDIM><<<NROWS / 128, 256, 0, stream>>>(
      ALoadPlain{bt, HDIM}, Wproj, EpiElu{bproj, big, HDIM});

  // 3. CSR-by-dst (batch independent), then atomic-free gather/accumulate
  hipMemsetAsync(tmp, 0, NBLK * sizeof(int), stream);
  k_hist<<<NEDGE / 256, 256, 0, stream>>>(edst, tmp);
  k_scan<<<1, 256, 0, stream>>>(tmp, offs);
  hipMemsetAsync(tmp, 0, NBLK * sizeof(int), stream);
  k_fill<<<NEDGE / 256, 256, 0, stream>>>(edst, offs, tmp, list);
  k_sortseg<<<NBLK / 256, 256, 0, stream>>>(offs, list);
  k_msg<<<dim3(NBLK / 8, BATCH), 256, 0, stream>>>(out_ew, big, esrc, offs, list,
                                                   msgs);

  // 4. x = bt + sigmoid([bt, msgs] @ W_gate + b_gate) * msgs
  gemm_kernel<2 * HDIM, HDIM><<<NROWS / 128, 256, 0, stream>>>(
      ALoadConcat{bt, msgs}, Wgate, EpiGate{bgate, bt, msgs, x});

  // 5. NL x (LN -> FFN with residual); last layer writes d_out directly
  for (int l = 0; l < NLAYER; ++l) {
    k_lnstats<<<NROWS / 8, 256, 0, stream>>>(x, stats);
    gemm_kernel<HDIM, 2 * HDIM><<<NROWS / 128, 256, 0, stream>>>(
        ALoadLN{x, stats, stats + NROWS, lnsc + (size_t)l * HDIM,
                lnbi + (size_t)l * HDIM},
        W1 + (size_t)l * HDIM * 2 * HDIM,
        EpiElu{b1 + (size_t)l * 2 * HDIM, big, 2 * HDIM});
    float* dst = (l == NLAYER - 1) ? out_x : x;
    gemm_kernel<2 * HDIM, HDIM><<<NROWS / 128, 256, 0, stream>>>(
        ALoadPlain{big, 2 * HDIM},
        W2 + (size_t)l * 2 * HDIM * HDIM,
        EpiRes{b2 + (size_t)l * HDIM, x, dst});
  }
}